// piecewise_maxpool_layer_8461085573367
// MI455X (gfx1250) — compile-verified
//
#include <hip/hip_runtime.h>
#include <stdint.h>

// Problem constants (match reference)
#define SEQ_LEN   256
#define FILTER_N  230
#define ROWS      16                         // sequence rows per LDS chunk
#define NCHUNK    (SEQ_LEN / ROWS)           // 16 chunks
#define CHUNK_BYTES (ROWS * FILTER_N * 4)    // 14720 B per chunk
#define COPY_PASSES 4                        // 4 passes * 256 lanes * 16 B = 16384 B
#define BUF_BYTES   (COPY_PASSES * 256 * 16) // 16384 B per buffer (chunk + padding)
#define NEG_INF   (-1e30f)

__global__ __launch_bounds__(256) void piecewise_maxpool_kernel(
    const float* __restrict__ conv,   // [B, S, F]
    const int*   __restrict__ pos,    // [B, 4]
    float*       __restrict__ out)    // [B, 3F]
{
    // Double-buffered staging area: 2 * 16 KB = 32 KB LDS, 16B aligned for b128.
    __shared__ __align__(16) float smem[2][BUF_BYTES / 4];

    const int b   = blockIdx.x;
    const int tid = threadIdx.x;

    // Piece boundaries are uniform per block -> SGPRs, scalar branches below.
    const int e1 = pos[b * 4 + 1];
    const int e2 = pos[b * 4 + 3];

    const char* gbase = (const char*)(conv + (size_t)b * SEQ_LEN * FILTER_N);

    // Raw LDS byte offsets of the two buffers (generic -> LDS addrspace cast).
    const uint32_t lds_base[2] = {
        (uint32_t)(uintptr_t)(__attribute__((address_space(3))) float*)&smem[0][0],
        (uint32_t)(uintptr_t)(__attribute__((address_space(3))) float*)&smem[1][0]
    };

    // Issue one chunk's async global->LDS copy: exactly COPY_PASSES wave-wide
    // b128 instructions per wave (uniform across all waves, EXEC always full),
    // so ASYNCcnt bookkeeping is exact. Overhang lanes clamp their global
    // address back into the chunk (junk lands in LDS padding, never read).
    auto issue_copy = [&](int chunk, uint32_t lbase) {
        const char* cbase = gbase + (size_t)chunk * CHUNK_BYTES;
#pragma unroll
        for (int p = 0; p < COPY_PASSES; ++p) {
            uint32_t byte_off = (uint32_t)(p * 256 + tid) * 16u;
            uint32_t g_off    = (byte_off < (uint32_t)CHUNK_BYTES) ? byte_off : 0u;
            uint64_t gaddr    = (uint64_t)(uintptr_t)(cbase + g_off);
            uint32_t laddr    = lbase + byte_off;
            asm volatile("global_load_async_to_lds_b128 %0, %1, off"
                         :: "v"(laddr), "v"(gaddr)
                         : "memory");
        }
    };

    float m1 = NEG_INF, m2 = NEG_INF, m3 = NEG_INF;

    // Prologue: start chunk 0.
    issue_copy(0, lds_base[0]);

    for (int c = 0; c < NCHUNK; ++c) {
        if (c + 1 < NCHUNK) {
            // Prefetch next chunk into the other buffer, then wait until the
            // 4 older async ops (chunk c) have landed. Async loads complete
            // in order, so asynccnt <= 4 == "chunk c is resident".
            issue_copy(c + 1, lds_base[(c + 1) & 1]);
            asm volatile("s_wait_asynccnt 4" ::: "memory");
        } else {
            asm volatile("s_wait_asynccnt 0" ::: "memory");
        }
        __syncthreads();  // all waves' portions of chunk c are now visible

        if (tid < FILTER_N) {
            const float* buf = smem[c & 1];
            const int sbase  = c * ROWS;
#pragma unroll
            for (int r = 0; r < ROWS; ++r) {
                float v = buf[r * FILTER_N + tid];
                int   s = sbase + r;          // uniform -> scalar branches
                if (s < e1)      m1 = fmaxf(m1, v);
                else if (s < e2) m2 = fmaxf(m2, v);
                else             m3 = fmaxf(m3, v);
            }
        }
        __syncthreads();  // readers done before this buffer is overwritten
    }

    if (tid < FILTER_N) {
        float* orow = out + (size_t)b * (3 * FILTER_N);
        orow[tid]                = m1;
        orow[FILTER_N + tid]     = m2;
        orow[2 * FILTER_N + tid] = m3;
    }
}

extern "C" void kernel_launch(void* const* d_in, const int* in_sizes, int n_in,
                              void* d_out, int out_size, void* d_ws, size_t ws_size,
                              hipStream_t stream) {
    const float* conv = (const float*)d_in[0];
    const int*   pos  = (const int*)d_in[1];
    float*       out  = (float*)d_out;

    const int batch = in_sizes[0] / (SEQ_LEN * FILTER_N);  // 2048
    piecewise_maxpool_kernel<<<batch, 256, 0, stream>>>(conv, pos, out);
}